// FusedScaleMaskSoftmax_8177617731585
// MI455X (gfx1250) — compile-verified
//
#include <hip/hip_runtime.h>
#include <cstdint>

// Fused scaled causal softmax over rows of 2048 fp32.
// [2,16,2048,2048] -> 65536 rows. HBM-bound: ~1 GiB traffic @ 23.3 TB/s ~= 46us.
// CDNA5 path: async global->LDS row prefetch (ASYNCcnt) + wave32 shfl reductions
// + non-temporal streaming stores.

#define THREADS      256
#define ROWS_PER_WG  8
#define SK           2048
#define NWAVES       (THREADS / 32)

typedef float v4f __attribute__((ext_vector_type(4)));
typedef int   v4i __attribute__((ext_vector_type(4)));

#if defined(__AMDGCN__) && __has_builtin(__builtin_amdgcn_global_load_async_to_lds_b128)
#define USE_ASYNC 1
#else
#define USE_ASYNC 0
#endif

// builtin signature (from clang diagnostic): (v4i AS1*, v4i AS3*, Imm, Imm)
#define AS1_V4I(p) ((__attribute__((address_space(1))) v4i*)(p))
#define AS3_V4I(p) ((__attribute__((address_space(3))) v4i*)(p))

template <int N>
__device__ __forceinline__ void wait_async() {
#if __has_builtin(__builtin_amdgcn_s_wait_asynccnt)
  __builtin_amdgcn_s_wait_asynccnt(N);
#else
  asm volatile("s_wait_asynccnt %0" ::"i"(N) : "memory");
#endif
}

__global__ __launch_bounds__(THREADS) void fused_causal_softmax_kernel(
    const float* __restrict__ in, float* __restrict__ out) {
#if USE_ASYNC
  __shared__ __align__(16) float buf[2][SK];  // double-buffered row staging (16 KB)
#endif
  __shared__ float red[NWAVES];

  const int t    = threadIdx.x;
  const int wid  = t >> 5;
  const int lane = t & 31;
  const long long row0 = (long long)blockIdx.x * ROWS_PER_WG;

  // thread t owns columns [c0, c0+3] and [c1, c1+3]
  const int c0 = t * 4;
  const int c1 = (SK / 2) + t * 4;

#if USE_ASYNC
  {
    // prologue: async-prefetch row 0 into buffer 0 (each lane fetches the
    // exact 16B it will later read -> only the wave's own asynccnt matters)
    const char* g = (const char*)(in + row0 * (long long)SK);
    __builtin_amdgcn_global_load_async_to_lds_b128(
        AS1_V4I(g + (size_t)c0 * 4), AS3_V4I(&buf[0][c0]), 0, 0);
    __builtin_amdgcn_global_load_async_to_lds_b128(
        AS1_V4I(g + (size_t)c1 * 4), AS3_V4I(&buf[0][c1]), 0, 0);
  }
#endif

  for (int r = 0; r < ROWS_PER_WG; ++r) {
    const long long row = row0 + r;
    const int i = (int)(row & (SK - 1));  // causal row index within 2048x2048 tile

    v4f x0, x1;
#if USE_ASYNC
    const int cur = r & 1;
    if (r + 1 < ROWS_PER_WG) {
      // issue prefetch of next row into the other buffer, then wait until only
      // those 2 ops are outstanding (in-order completion => current row ready)
      const char* gn = (const char*)(in + (row + 1) * (long long)SK);
      __builtin_amdgcn_global_load_async_to_lds_b128(
          AS1_V4I(gn + (size_t)c0 * 4), AS3_V4I(&buf[cur ^ 1][c0]), 0, 0);
      __builtin_amdgcn_global_load_async_to_lds_b128(
          AS1_V4I(gn + (size_t)c1 * 4), AS3_V4I(&buf[cur ^ 1][c1]), 0, 0);
      wait_async<2>();
    } else {
      wait_async<0>();
    }
    asm volatile("" ::: "memory");
    x0 = *(const v4f*)(&buf[cur][c0]);
    x1 = *(const v4f*)(&buf[cur][c1]);
#else
    const float* grow = in + row * (long long)SK;
    x0 = __builtin_nontemporal_load((const v4f*)(grow + c0));
    x1 = __builtin_nontemporal_load((const v4f*)(grow + c1));
#endif

    float xv[8];
#pragma unroll
    for (int k = 0; k < 4; ++k) {
      xv[k]     = x0[k] * 0.125f;
      xv[4 + k] = x1[k] * 0.125f;
    }

    // ---- masked row max ----
    float m = -INFINITY;
#pragma unroll
    for (int k = 0; k < 8; ++k) {
      const int col = (k < 4) ? (c0 + k) : (c1 + k - 4);
      m = (col <= i) ? fmaxf(m, xv[k]) : m;
    }
#pragma unroll
    for (int off = 16; off > 0; off >>= 1)  // wave32 reduction
      m = fmaxf(m, __shfl_xor(m, off, 32));
    if (lane == 0) red[wid] = m;
    __syncthreads();
    float mrow = red[0];
#pragma unroll
    for (int w = 1; w < NWAVES; ++w) mrow = fmaxf(mrow, red[w]);
    __syncthreads();

    // ---- exp + masked row sum ----
    float p[8];
    float s = 0.f;
#pragma unroll
    for (int k = 0; k < 8; ++k) {
      const int col = (k < 4) ? (c0 + k) : (c1 + k - 4);
      const float e = __expf(xv[k] - mrow);
      p[k] = (col <= i) ? e : 0.f;  // masked positions -> exactly 0
      s += p[k];
    }
#pragma unroll
    for (int off = 16; off > 0; off >>= 1) s += __shfl_xor(s, off, 32);
    if (lane == 0) red[wid] = s;
    __syncthreads();
    float srow = 0.f;
#pragma unroll
    for (int w = 0; w < NWAVES; ++w) srow += red[w];
    __syncthreads();

    // ---- normalize + streaming store ----
    const float inv = 1.0f / srow;  // srow >= 1 always (diagonal element)
    float* orow = out + row * (long long)SK;
    v4f o0, o1;
#pragma unroll
    for (int k = 0; k < 4; ++k) {
      o0[k] = p[k] * inv;
      o1[k] = p[4 + k] * inv;
    }
    __builtin_nontemporal_store(o0, (v4f*)(orow + c0));
    __builtin_nontemporal_store(o1, (v4f*)(orow + c1));
  }
}

extern "C" void kernel_launch(void* const* d_in, const int* in_sizes, int n_in,
                              void* d_out, int out_size, void* d_ws, size_t ws_size,
                              hipStream_t stream) {
  (void)n_in; (void)d_ws; (void)ws_size; (void)out_size;
  const float* in = (const float*)d_in[0];
  float* out = (float*)d_out;
  const long long total = (long long)in_sizes[0];      // 2*16*2048*2048
  const long long rows  = total / SK;                  // 65536
  const int blocks = (int)(rows / ROWS_PER_WG);        // 8192
  fused_causal_softmax_kernel<<<blocks, THREADS, 0, stream>>>(in, out);
}